// NodeClassificationMPNSimpleWithRef_60627758350345
// MI455X (gfx1250) — compile-verified
//
#include <hip/hip_runtime.h>
#include <hip/hip_bf16.h>

typedef __attribute__((ext_vector_type(2))) float v2f;
typedef __attribute__((ext_vector_type(8))) float v8f;

#define GEMM_BLOCK 128
#define GEMM_WAVES 4
#define WAVE_M 32                 // 2 row-sets of 16 per wave
#define TILE_M (GEMM_WAVES * WAVE_M)   // 128 rows per block
#define TILE_N 64

// A-operand described as up to 4 concatenated row-segments.
// mode: 0 = identity (segment row = output row), 1 = row[], 2 = col[]
struct Segs {
  const float* p0; const float* p1; const float* p2; const float* p3;
  int w0, w1, w2, w3;
  int md0, md1, md2, md3;
  int nseg;
};

// out[M,N] = act(A[M,K] @ W[K,N] + bias),  A gathered via Segs.
// Each wave computes a 32x64 tile with V_WMMA_F32_16X16X4_F32 (8 wmma / k-step).
__global__ __launch_bounds__(GEMM_BLOCK)
void gemm_seg_wmma(Segs S, const int* __restrict__ rowIdx, const int* __restrict__ colIdx,
                   const float* __restrict__ W, const float* __restrict__ bias,
                   float* __restrict__ out, int M, int K, int N, int doRelu)
{
  const int wave  = threadIdx.x >> 5;
  const int lane  = threadIdx.x & 31;
  const int mBase = blockIdx.y * TILE_M + wave * WAVE_M;
  const int n0    = blockIdx.x * TILE_N;
  if (mBase >= M) return;              // wave-uniform: EXEC stays all-ones for WMMA

  const int lr = lane & 15;            // column lane within half-wave
  const int hi = lane >> 4;            // 0: K=k..k+1, 1: K=k+2..k+3 (A/B VGPR layout)

  // two A row-sets per wave (clamped; out-of-range rows never stored)
  const int r0 = ((mBase + lr)      < M) ? (mBase + lr)      : (M - 1);
  const int r1 = ((mBase + 16 + lr) < M) ? (mBase + 16 + lr) : (M - 1);

  auto rowbase = [&](const float* b, int w, int mode, int sr) -> const float* {
    int idx = sr;
    if (mode == 1)      idx = rowIdx[sr];
    else if (mode == 2) idx = colIdx[sr];
    return b + (size_t)idx * (size_t)w;
  };
  const float* s0a = rowbase(S.p0, S.w0, S.md0, r0);
  const float* s0b = rowbase(S.p0, S.w0, S.md0, r1);
  const float* s1a = (S.nseg > 1) ? rowbase(S.p1, S.w1, S.md1, r0) : s0a;
  const float* s1b = (S.nseg > 1) ? rowbase(S.p1, S.w1, S.md1, r1) : s0b;
  const float* s2a = (S.nseg > 2) ? rowbase(S.p2, S.w2, S.md2, r0) : s0a;
  const float* s2b = (S.nseg > 2) ? rowbase(S.p2, S.w2, S.md2, r1) : s0b;
  const float* s3a = (S.nseg > 3) ? rowbase(S.p3, S.w3, S.md3, r0) : s0a;
  const float* s3b = (S.nseg > 3) ? rowbase(S.p3, S.w3, S.md3, r1) : s0b;

  // clamped B/output columns per N-tile (B column n only feeds output column n,
  // and out-of-range output columns are never stored -> no zeroing needed)
  const int nr0 = n0 + lr, nr1 = n0 + 16 + lr, nr2 = n0 + 32 + lr, nr3 = n0 + 48 + lr;
  const int nc0 = (nr0 < N) ? nr0 : (N - 1);
  const int nc1 = (nr1 < N) ? nr1 : (N - 1);
  const int nc2 = (nr2 < N) ? nr2 : (N - 1);
  const int nc3 = (nr3 < N) ? nr3 : (N - 1);

  v8f a00 = {0.f,0.f,0.f,0.f,0.f,0.f,0.f,0.f};
  v8f a01 = a00, a02 = a00, a03 = a00;   // row-set 0, tiles 0..3
  v8f a10 = a00, a11 = a00, a12 = a00, a13 = a00;  // row-set 1

  // W row pointer walks the whole K range across segments
  const float* wrow = W + (size_t)(2 * hi) * (size_t)N;

#pragma unroll
  for (int s = 0; s < 4; ++s) {
    if (s < S.nseg) {                                   // uniform branch
      const int wseg      = (s == 0) ? S.w0 : (s == 1) ? S.w1 : (s == 2) ? S.w2 : S.w3;
      const float* spA    = (s == 0) ? s0a  : (s == 1) ? s1a  : (s == 2) ? s2a  : s3a;
      const float* spB    = (s == 0) ? s0b  : (s == 1) ? s1b  : (s == 2) ? s2b  : s3b;
      for (int kk = 0; kk < wseg; kk += 4) {
        const v2f af0 = *(const v2f*)(spA + kk + 2 * hi);
        const v2f af1 = *(const v2f*)(spB + kk + 2 * hi);
        const float* wp0 = wrow;
        const float* wp1 = wrow + N;
#define WMMA_NT2(ACCA, ACCB, NC)                                                 \
        { v2f b; b[0] = wp0[NC]; b[1] = wp1[NC];                                 \
          ACCA = __builtin_amdgcn_wmma_f32_16x16x4_f32(false, af0, false, b,     \
                                                       (short)0, ACCA, false, false); \
          ACCB = __builtin_amdgcn_wmma_f32_16x16x4_f32(false, af1, false, b,     \
                                                       (short)0, ACCB, false, false); }
        WMMA_NT2(a00, a10, nc0)
        WMMA_NT2(a01, a11, nc1)
        WMMA_NT2(a02, a12, nc2)
        WMMA_NT2(a03, a13, nc3)
#undef WMMA_NT2
        wrow += 4 * (size_t)N;
      }
    }
  }

  // C/D layout: VGPR j -> M = base + j (lanes 0-15) / base + j + 8 (lanes 16-31), N per lane lr
#pragma unroll
  for (int j = 0; j < 8; ++j) {
    const int m0 = mBase + j + 8 * hi;
    const int m1 = m0 + 16;
#define STORE_NT(ACC, M_, NR, NC)                                                \
    { if ((M_) < M && (NR) < N) {                                                \
        float v = ACC[j] + bias[NC];                                             \
        if (doRelu) v = fmaxf(v, 0.f);                                           \
        out[(size_t)(M_) * (size_t)N + (NR)] = v; } }
    STORE_NT(a00, m0, nr0, nc0)
    STORE_NT(a01, m0, nr1, nc1)
    STORE_NT(a02, m0, nr2, nc2)
    STORE_NT(a03, m0, nr3, nc3)
    STORE_NT(a10, m1, nr0, nc0)
    STORE_NT(a11, m1, nr1, nc1)
    STORE_NT(a12, m1, nr2, nc2)
    STORE_NT(a13, m1, nr3, nc3)
#undef STORE_NT
  }
}

__global__ void concat2_kernel(float* __restrict__ dst,
                               const float* __restrict__ a, int wa,
                               const float* __restrict__ b, int wb, size_t rows)
{
  const int w = wa + wb;
  const size_t total = rows * (size_t)w;
  for (size_t i = (size_t)blockIdx.x * blockDim.x + threadIdx.x; i < total;
       i += (size_t)gridDim.x * blockDim.x) {
    const size_t r = i / (size_t)w;
    const int    c = (int)(i - r * (size_t)w);
    dst[i] = (c < wa) ? a[r * (size_t)wa + c] : b[r * (size_t)wb + (c - wa)];
  }
}

__global__ void zero_kernel(float* __restrict__ p, size_t n)
{
  for (size_t i = (size_t)blockIdx.x * blockDim.x + threadIdx.x; i < n;
       i += (size_t)gridDim.x * blockDim.x)
    p[i] = 0.f;
}

__global__ void scatter_add_kernel(float* __restrict__ agg,
                                   const float* __restrict__ e,
                                   const int* __restrict__ colIdx,
                                   size_t E, int W)
{
  const size_t total = E * (size_t)W;
  for (size_t i = (size_t)blockIdx.x * blockDim.x + threadIdx.x; i < total;
       i += (size_t)gridDim.x * blockDim.x) {
    const size_t r = i / (size_t)W;
    const int    c = (int)(i - r * (size_t)W);
    unsafeAtomicAdd(&agg[(size_t)colIdx[r] * (size_t)W + c], e[i]);   // global_atomic_add_f32
  }
}

__global__ void gemv_kernel(const float* __restrict__ A, const float* __restrict__ w,
                            const float* __restrict__ b, float* __restrict__ out,
                            int M, int K)
{
  const int i = blockIdx.x * blockDim.x + threadIdx.x;
  if (i >= M) return;
  const float* a = A + (size_t)i * (size_t)K;
  float s = b[0];
  for (int k = 0; k < K; ++k) s += a[k] * w[k];
  out[i] = s;
}

// ---------------- host side ----------------

// Input layout: setup_inputs() insertion order, params leaves in insertion order.
enum {
  IN_X = 0, IN_EATTR = 1, IN_EIDX = 2,
  P_NE_W0 = 3,  P_NE_B0, P_NE_W1, P_NE_B1,          // node_emb
  P_EE_W0 = 7,  P_EE_B0, P_EE_W1, P_EE_B1,          // edge_emb
  P_ME_EW0 = 11, P_ME_EB0, P_ME_EW1, P_ME_EB1,      // mpn_edge.edge_mlp
  P_ME_NW = 15,  P_ME_NB,                           // mpn_edge.node_mlp
  P_MN_EW0 = 17, P_MN_EB0, P_MN_EW1, P_MN_EB1,      // mpn_node.edge_mlp
  P_MN_NW = 21,  P_MN_NB,                           // mpn_node.node_mlp
  P_EO_W = 23,   P_EO_B,                            // edge_out
  P_EC_W = 25,   P_EC_B,                            // edge_cls
  P_NC_W0 = 27,  P_NC_B0, P_NC_W1, P_NC_B1,         // node_cls
  P_CL_W0 = 31,  P_CL_B0, P_CL_W1, P_CL_B1          // cls
};

static inline Segs mk1(const float* a, int wa) {
  Segs s{}; s.p0 = a; s.w0 = wa; s.md0 = 0; s.nseg = 1; return s;
}
static inline Segs mk2(const float* a, int wa, const float* b, int wb) {
  Segs s{}; s.p0 = a; s.w0 = wa; s.md0 = 0;
  s.p1 = b; s.w1 = wb; s.md1 = 0; s.nseg = 2; return s;
}
static inline Segs mk4(const float* a, int wa, int ma, const float* b, int wb, int mb,
                       const float* c, int wc, int mc, const float* d, int wd, int md) {
  Segs s{}; s.p0=a; s.w0=wa; s.md0=ma; s.p1=b; s.w1=wb; s.md1=mb;
  s.p2=c; s.w2=wc; s.md2=mc; s.p3=d; s.w3=wd; s.md3=md; s.nseg = 4; return s;
}

static inline void run_gemm(hipStream_t st, const Segs& S, const int* row, const int* col,
                            const float* W, const float* b, float* out,
                            int M, int K, int N, int relu)
{
  dim3 grid((unsigned)((N + TILE_N - 1) / TILE_N),
            (unsigned)((M + TILE_M - 1) / TILE_M), 1);
  gemm_seg_wmma<<<grid, GEMM_BLOCK, 0, st>>>(S, row, col, W, b, out, M, K, N, relu);
}

static inline unsigned nblk(size_t total, unsigned tpb) {
  size_t b = (total + tpb - 1) / tpb;
  if (b > 262144) b = 262144;
  if (b == 0) b = 1;
  return (unsigned)b;
}

extern "C" void kernel_launch(void* const* d_in, const int* in_sizes, int n_in,
                              void* d_out, int out_size, void* d_ws, size_t ws_size,
                              hipStream_t stream)
{
  (void)n_in; (void)out_size; (void)ws_size;
  const int Nn = in_sizes[IN_X]     / 64;   // 20000 nodes
  const int Ne = in_sizes[IN_EATTR] / 32;   // 480000 edges

  const float* x   = (const float*)d_in[IN_X];
  const float* ea  = (const float*)d_in[IN_EATTR];
  const int*   eix = (const int*)d_in[IN_EIDX];
  const int* row = eix;
  const int* col = eix + Ne;
  auto P = [&](int i) { return (const float*)d_in[i]; };

  float* outv       = (float*)d_out;
  float* pred_edge  = outv;                 // [E]
  float* pred_node  = outv + Ne;            // [N]
  float* pred_class = outv + Ne + (size_t)Nn;   // [N,17]

  // ---- carve workspace (256B aligned) ----
  char*  base = (char*)d_ws;
  size_t off  = 0;
  auto carve = [&](size_t nflt) -> float* {
    float* p = (float*)(base + off);
    off += nflt * 4;
    off = (off + 255) & ~(size_t)255;
    return p;
  };
  float* h      = carve((size_t)Ne * 256);  // edge-MLP hidden
  float* efe    = carve((size_t)Ne * 128);  // ef0 (edge phase) / embedded edge feats
  float* efa    = carve((size_t)Ne * 128);  // updated ef (edge phase)
  float* efo    = carve((size_t)Ne * 32);   // ef_out
  float* ef1600 = carve((size_t)Ne * 160);  // ef0 (node phase)
  float* nf     = carve((size_t)Nn * 128);
  float* nf0    = carve((size_t)Nn * 128);
  float* nfc    = carve((size_t)Nn * 256);
  float* agg    = carve((size_t)Nn * 160);
  float* tA     = carve((size_t)Nn * 128);
  float* efa160 = efe;   // alias: [efe..efa] (E*256 floats) retired before node phase

  // ---- embeddings ----
  run_gemm(stream, mk1(x, 64),   row, col, P(P_NE_W0), P(P_NE_B0), tA,  Nn, 64,  128, 1);
  run_gemm(stream, mk1(tA, 128), row, col, P(P_NE_W1), P(P_NE_B1), nf,  Nn, 128, 128, 0);
  run_gemm(stream, mk1(ea, 32),  row, col, P(P_EE_W0), P(P_EE_B0), h,   Ne, 32,  128, 1);
  run_gemm(stream, mk1(h, 128),  row, col, P(P_EE_W1), P(P_EE_B1), efe, Ne, 128, 128, 0);
  hipMemcpyAsync(nf0, nf, (size_t)Nn * 128 * 4, hipMemcpyDeviceToDevice, stream);

  // ---- 4 edge message-passing steps ----
  const float* efc = efe;    // current ef (starts == ef0)
  for (int it = 0; it < 4; ++it) {
    concat2_kernel<<<nblk((size_t)Nn * 256, 256), 256, 0, stream>>>(nfc, nf0, 128, nf, 128, (size_t)Nn);
    // h = relu([nfc[row] | nfc[col] | ef0 | ef] @ W0 + b0)   K=768, N=256
    run_gemm(stream, mk4(nfc, 256, 1, nfc, 256, 2, efe, 128, 0, efc, 128, 0),
             row, col, P(P_ME_EW0), P(P_ME_EB0), h, Ne, 768, 256, 1);
    // e_new = relu(h @ W1 + b1)   K=256, N=128
    run_gemm(stream, mk1(h, 256), row, col, P(P_ME_EW1), P(P_ME_EB1), efa, Ne, 256, 128, 1);
    efc = efa;
    // agg = segment_sum(e_new, col)
    zero_kernel<<<nblk((size_t)Nn * 128, 256), 256, 0, stream>>>(agg, (size_t)Nn * 128);
    scatter_add_kernel<<<nblk((size_t)Ne * 128, 256), 256, 0, stream>>>(agg, efa, col, (size_t)Ne, 128);
    // nf = relu([nfc | agg] @ Wn + bn)   K=384, N=128
    run_gemm(stream, mk2(nfc, 256, agg, 128), row, col, P(P_ME_NW), P(P_ME_NB), nf, Nn, 384, 128, 1);
  }

  // ---- edge head ----
  run_gemm(stream, mk1(efc, 128), row, col, P(P_EO_W), P(P_EO_B), efo, Ne, 128, 32, 1);
  gemv_kernel<<<(Ne + 255) / 256, 256, 0, stream>>>(efo, P(P_EC_W), P(P_EC_B), pred_edge, Ne, 32);

  // ---- 2 node message-passing steps (edge width 160) ----
  hipMemcpyAsync(nf0, nf, (size_t)Nn * 128 * 4, hipMemcpyDeviceToDevice, stream);
  concat2_kernel<<<nblk((size_t)Ne * 160, 256), 256, 0, stream>>>(ef1600, efo, 32, efc, 128, (size_t)Ne);
  const float* efc160 = ef1600;
  for (int it = 0; it < 2; ++it) {
    concat2_kernel<<<nblk((size_t)Nn * 256, 256), 256, 0, stream>>>(nfc, nf0, 128, nf, 128, (size_t)Nn);
    // h = relu([nfc[row] | nfc[col] | ef0 | ef] @ W0 + b0)   K=832, N=256
    run_gemm(stream, mk4(nfc, 256, 1, nfc, 256, 2, ef1600, 160, 0, efc160, 160, 0),
             row, col, P(P_MN_EW0), P(P_MN_EB0), h, Ne, 832, 256, 1);
    // e_new = relu(h @ W1 + b1)   K=256, N=160
    run_gemm(stream, mk1(h, 256), row, col, P(P_MN_EW1), P(P_MN_EB1), efa160, Ne, 256, 160, 1);
    efc160 = efa160;
    zero_kernel<<<nblk((size_t)Nn * 160, 256), 256, 0, stream>>>(agg, (size_t)Nn * 160);
    scatter_add_kernel<<<nblk((size_t)Ne * 160, 256), 256, 0, stream>>>(agg, efa160, col, (size_t)Ne, 160);
    // nf = relu([nfc | agg] @ Wn + bn)   K=416, N=128
    run_gemm(stream, mk2(nfc, 256, agg, 160), row, col, P(P_MN_NW), P(P_MN_NB), nf, Nn, 416, 128, 1);
  }

  // ---- node heads ----
  run_gemm(stream, mk1(nf, 128), row, col, P(P_NC_W0), P(P_NC_B0), tA, Nn, 128, 128, 1);
  gemv_kernel<<<(Nn + 255) / 256, 256, 0, stream>>>(tA, P(P_NC_W1), P(P_NC_B1), pred_node, Nn, 128);
  run_gemm(stream, mk1(nf, 128), row, col, P(P_CL_W0), P(P_CL_B0), tA, Nn, 128, 128, 1);
  run_gemm(stream, mk1(tA, 128), row, col, P(P_CL_W1), P(P_CL_B1), pred_class, Nn, 128, 17, 0);
}